// MSHGEncoder_23682449670475
// MI455X (gfx1250) — compile-verified
//
#include <hip/hip_runtime.h>
#include <hip/hip_bf16.h>
#include <float.h>

// ---------------- problem constants (fixed by the reference) ----------------
constexpr int N_NODES = 100000;
constexpr int E_EDGES = 1000000;
constexpr int R_REL   = 2;
constexpr int DIN     = 128;
constexpr int H_HEADS = 4;
constexpr int D_HID   = 32;
constexpr int HD      = H_HEADS * D_HID;   // 128
constexpr int REL_IN  = 64;
constexpr int REL_H   = 32;
constexpr float NEG_SLOPE = 0.2f;

typedef __attribute__((ext_vector_type(2))) float v2f;
typedef __attribute__((ext_vector_type(8))) float v8f;

// ---------------- helpers ----------------
__device__ __forceinline__ void atomicMaxF(float* addr, float val) {
    // float-ordered max via integer atomics (no CAS loop).
    if (val >= 0.0f) {
        atomicMax((int*)addr, __float_as_int(val));
    } else {
        atomicMin((unsigned int*)addr, (unsigned int)__float_as_int(val));
    }
}

__device__ __forceinline__ float leakyf(float v) {
    return v >= 0.0f ? v : NEG_SLOPE * v;
}

// ---------------- kernels ----------------
__global__ void fill_f32(float* __restrict__ p, float v, long long n) {
    long long i = (long long)blockIdx.x * blockDim.x + threadIdx.x;
    if (i < n) p[i] = v;
}

__global__ void degree_kernel(const int* __restrict__ src, const int* __restrict__ dst,
                              float* __restrict__ deg_out, float* __restrict__ deg_in,
                              int total) {
    int i = blockIdx.x * blockDim.x + threadIdx.x;
    if (i < total) {
        atomicAdd(&deg_out[src[i]], 1.0f);
        atomicAdd(&deg_in[dst[i]], 1.0f);
    }
}

// GraphConv scatter: hg[dst] += x[src] * rsqrt(max(deg_out[src],1))
__global__ void gcn_scatter(const float* __restrict__ x, const int* __restrict__ src,
                            const int* __restrict__ dst, const float* __restrict__ deg_out,
                            float* __restrict__ hg, int total_edges) {
    long long t = (long long)blockIdx.x * blockDim.x + threadIdx.x;
    int f = (int)(t & 127);
    long long e = t >> 7;
    if (e >= total_edges) return;
    int s = src[e], d = dst[e];
    float dg = deg_out[s];
    dg = dg < 1.0f ? 1.0f : dg;
    atomicAdd(&hg[(size_t)d * HD + f], x[(size_t)s * DIN + f] * rsqrtf(dg));
}

// C[n,:] = act( rowscale(n) * A[n,:] @ W + bias ),  A: n_rows x 128, W: 128 x 128.
// One wave -> one 16x16 tile via V_WMMA_F32_16X16X4_F32; 8 waves/block cover 128 cols.
__global__ __launch_bounds__(256)
void gemm128_wmma(const float* __restrict__ A, const float* __restrict__ W,
                  const float* __restrict__ bias, const float* __restrict__ deg_rowscale,
                  float* __restrict__ out, int relu_flag) {
    const int wave = threadIdx.x >> 5;
    const int lane = threadIdx.x & 31;
    const int m0   = blockIdx.x * 16;

    const int mrow = lane & 15;          // A-row within tile (both lane halves)
    const int koff = (lane >> 4) * 2;    // lanes 0-15 -> K 0,1 ; lanes 16-31 -> K 2,3
    const int ncol = wave * 16 + (lane & 15);

    float rs = 1.0f;
    if (deg_rowscale) {
        float dg = deg_rowscale[m0 + mrow];
        dg = dg < 1.0f ? 1.0f : dg;
        rs = rsqrtf(dg);
    }
    const float* __restrict__ Arow = A + (size_t)(m0 + mrow) * DIN;

    v8f c = {};
    #pragma unroll
    for (int k0 = 0; k0 < DIN; k0 += 4) {
        v2f a, b;
        a.x = Arow[k0 + koff] * rs;
        a.y = Arow[k0 + koff + 1] * rs;
        b.x = W[(size_t)(k0 + koff) * HD + ncol];
        b.y = W[(size_t)(k0 + koff + 1) * HD + ncol];
        c = __builtin_amdgcn_wmma_f32_16x16x4_f32(
                /*neg_a=*/false, a, /*neg_b=*/false, b,
                /*c_mod=*/(short)0, c, /*reuse_a=*/false, /*reuse_b=*/false);
    }

    const int half = lane >> 4;   // C layout: lanes 0-15 M=v, lanes 16-31 M=8+v
    const float bv = bias[ncol];
    #pragma unroll
    for (int v = 0; v < 8; ++v) {
        int row = m0 + v + half * 8;
        float val = c[v] + bv;
        if (relu_flag) val = fmaxf(val, 0.0f);
        out[(size_t)row * HD + ncol] = val;
    }
}

// tiny relation-level tensors: attn (R,H,2,D) and rel_feat (R,H,D). single block.
__global__ void small_precompute(const float* __restrict__ rel_emb, const float* __restrict__ W_relT,
                                 const float* __restrict__ W_relprop, const float* __restrict__ b_relprop,
                                 const float* __restrict__ W_fuse,
                                 float* __restrict__ attn, float* __restrict__ rel_feat) {
    __shared__ float sprop[R_REL * H_HEADS * REL_H];  // 256
    const int tid = threadIdx.x;

    // attn[r][j], j in [0,256): attn = rel_emb[r] @ W_relT[r]
    for (int idx = tid; idx < R_REL * 256; idx += 256) {
        int r = idx >> 8, j = idx & 255;
        float acc = 0.0f;
        for (int k = 0; k < REL_IN; ++k)
            acc += rel_emb[r * REL_IN + k] * W_relT[((size_t)r * REL_IN + k) * 256 + j];
        attn[idx] = acc;
    }
    // rel_prop[r][m], m in [0,128)
    for (int idx = tid; idx < R_REL * 128; idx += 256) {
        int r = idx >> 7, m = idx & 127;
        float acc = b_relprop[idx];
        for (int k = 0; k < REL_IN; ++k)
            acc += rel_emb[r * REL_IN + k] * W_relprop[((size_t)r * REL_IN + k) * 128 + m];
        sprop[idx] = acc;
    }
    __syncthreads();
    // rel_feat[r,h,d] = sum_k sprop[r,h*32+k] * W_fuse[r,h,k,d]   (exactly 256 outputs)
    {
        int idx = tid;
        int d = idx & 31, h = (idx >> 5) & 3, r = idx >> 7;
        float acc = 0.0f;
        for (int k = 0; k < REL_H; ++k)
            acc += sprop[r * 128 + h * REL_H + k] *
                   W_fuse[(((size_t)(r * H_HEADS + h)) * REL_H + k) * D_HID + d];
        rel_feat[idx] = acc;
    }
}

// es[r,n,h] = <feat_src[r][n,h,:], attn[r,h,0,:]> ; ed likewise with feat_dst, attn[r,h,1,:]
__global__ void node_scores(const float* __restrict__ feat_src, const float* __restrict__ feat_dst,
                            const float* __restrict__ attn,
                            float* __restrict__ es, float* __restrict__ ed) {
    int t = blockIdx.x * blockDim.x + threadIdx.x;
    if (t >= R_REL * N_NODES * H_HEADS) return;
    int h = t & 3;
    int n = (t >> 2) % N_NODES;
    int r = t / (N_NODES * H_HEADS);
    const float* fs = feat_src + ((size_t)r * N_NODES + n) * HD + h * D_HID;
    const float* fd = feat_dst + (size_t)n * HD + h * D_HID;
    const float* a0 = attn + r * 512 + h * 64;       // attn[r,h,0,:]
    const float* a1 = a0 + 32;                        // attn[r,h,1,:]
    float s0 = 0.0f, s1 = 0.0f;
    #pragma unroll
    for (int d = 0; d < D_HID; ++d) { s0 += fs[d] * a0[d]; s1 += fd[d] * a1[d]; }
    es[t] = s0;
    ed[t] = s1;
}

// pass 1: e = leaky(es[src]+ed[dst]); segment max into mbuf
__global__ void edge_max(const int* __restrict__ src, const int* __restrict__ dst,
                         const float* __restrict__ es, const float* __restrict__ ed,
                         float* __restrict__ ebuf, float* __restrict__ mbuf) {
    int t = blockIdx.x * blockDim.x + threadIdx.x;
    if (t >= R_REL * E_EDGES) return;
    int r = t / E_EDGES;
    int s = src[t], d = dst[t];
    #pragma unroll
    for (int h = 0; h < H_HEADS; ++h) {
        float e = leakyf(es[((size_t)r * N_NODES + s) * H_HEADS + h] +
                         ed[((size_t)r * N_NODES + d) * H_HEADS + h]);
        ebuf[(size_t)t * H_HEADS + h] = e;
        atomicMaxF(&mbuf[((size_t)r * N_NODES + d) * H_HEADS + h], e);
    }
}

// pass 2: ex = exp(e - m[dst]); segment sum into sbuf; overwrite ebuf with ex
__global__ void edge_expsum(const int* __restrict__ dst,
                            const float* __restrict__ mbuf,
                            float* __restrict__ ebuf, float* __restrict__ sbuf) {
    int t = blockIdx.x * blockDim.x + threadIdx.x;
    if (t >= R_REL * E_EDGES) return;
    int r = t / E_EDGES;
    int d = dst[t];
    #pragma unroll
    for (int h = 0; h < H_HEADS; ++h) {
        float ex = __expf(ebuf[(size_t)t * H_HEADS + h] -
                          mbuf[((size_t)r * N_NODES + d) * H_HEADS + h]);
        ebuf[(size_t)t * H_HEADS + h] = ex;
        atomicAdd(&sbuf[((size_t)r * N_NODES + d) * H_HEADS + h], ex);
    }
}

// pass 3: agg[dst,h,:] += (ex / (s[dst,h]+1e-9)) * feat_src[src,h,:]
__global__ void edge_agg(const int* __restrict__ src, const int* __restrict__ dst,
                         const float* __restrict__ ebuf, const float* __restrict__ sbuf,
                         const float* __restrict__ feat_src, float* __restrict__ agg) {
    long long t = (long long)blockIdx.x * blockDim.x + threadIdx.x;
    int c = (int)(t & 127);
    long long ei = t >> 7;
    if (ei >= (long long)R_REL * E_EDGES) return;
    int r = (int)(ei / E_EDGES);
    int h = c >> 5;
    int s = src[ei], d = dst[ei];
    float alpha = ebuf[(size_t)ei * H_HEADS + h] /
                  (sbuf[((size_t)r * N_NODES + d) * H_HEADS + h] + 1e-9f);
    atomicAdd(&agg[((size_t)r * N_NODES + d) * HD + c],
              alpha * feat_src[((size_t)r * N_NODES + s) * HD + c]);
}

// scores[r,n,h] = leaky( <relu(agg[r,n,h,:]), rel_feat[r,h,:]> )
__global__ void rel_scores(const float* __restrict__ agg, const float* __restrict__ rel_feat,
                           float* __restrict__ scores) {
    int t = blockIdx.x * blockDim.x + threadIdx.x;
    if (t >= R_REL * N_NODES * H_HEADS) return;
    int h = t & 3;
    int n = (t >> 2) % N_NODES;
    int r = t / (N_NODES * H_HEADS);
    const float* a  = agg + ((size_t)r * N_NODES + n) * HD + h * D_HID;
    const float* rf = rel_feat + r * HD + h * D_HID;
    float acc = 0.0f;
    #pragma unroll
    for (int d = 0; d < D_HID; ++d) acc += fmaxf(a[d], 0.0f) * rf[d];
    scores[t] = leakyf(acc);
}

// softmax over relations, fuse, residual blend, final gate
__global__ void final_fuse(const float* __restrict__ agg, const float* __restrict__ scores,
                           const float* __restrict__ feat_dst, const float* __restrict__ node_level,
                           const float* __restrict__ residual_w, const float* __restrict__ scale_w,
                           float* __restrict__ out) {
    long long t = (long long)blockIdx.x * blockDim.x + threadIdx.x;
    if (t >= (long long)N_NODES * HD) return;
    int c = (int)(t & 127);
    int n = (int)(t >> 7);
    int h = c >> 5;
    float s0 = scores[((size_t)0 * N_NODES + n) * H_HEADS + h];
    float s1 = scores[((size_t)1 * N_NODES + n) * H_HEADS + h];
    float mx = fmaxf(s0, s1);
    float w0 = __expf(s0 - mx), w1 = __expf(s1 - mx);
    float inv = 1.0f / (w0 + w1);
    float f0 = fmaxf(agg[((size_t)0 * N_NODES + n) * HD + c], 0.0f);
    float f1 = fmaxf(agg[((size_t)1 * N_NODES + n) * HD + c], 0.0f);
    float fused = (w0 * f0 + w1 * f1) * inv;
    float beta = 1.0f / (1.0f + __expf(-residual_w[0]));
    fused = beta * fused + (1.0f - beta) * feat_dst[(size_t)n * HD + c];
    float a = 1.0f / (1.0f + __expf(-scale_w[0]));
    out[t] = fused * a + node_level[(size_t)n * HD + c] * (1.0f - a);
}

// ---------------- launch ----------------
static inline int cdiv_ll(long long a, int b) { return (int)((a + b - 1) / b); }

extern "C" void kernel_launch(void* const* d_in, const int* in_sizes, int n_in,
                              void* d_out, int out_size, void* d_ws, size_t ws_size,
                              hipStream_t stream) {
    const float* x          = (const float*)d_in[0];
    const int*   src        = (const int*)d_in[1];   // (R,E) flat
    const int*   dst        = (const int*)d_in[2];
    const float* rel_emb    = (const float*)d_in[3];
    const float* W_node     = (const float*)d_in[4];
    const float* b_node     = (const float*)d_in[5];
    const float* W_src      = (const float*)d_in[6];
    const float* b_src      = (const float*)d_in[7];
    const float* W_relT     = (const float*)d_in[8];
    const float* W_relprop  = (const float*)d_in[9];
    const float* b_relprop  = (const float*)d_in[10];
    const float* W_fuse     = (const float*)d_in[11];
    const float* residual_w = (const float*)d_in[12];
    const float* scale_w    = (const float*)d_in[13];
    const float* W_g        = (const float*)d_in[14];
    const float* b_g        = (const float*)d_in[15];
    float* out = (float*)d_out;

    // ---- workspace carve-up (floats) ----
    float* ws = (float*)d_ws;
    size_t o = 0;
    float* deg_out    = ws + o; o += N_NODES;
    float* deg_in     = ws + o; o += N_NODES;
    float* hg         = ws + o; o += (size_t)N_NODES * HD;
    float* node_level = ws + o; o += (size_t)N_NODES * HD;
    float* feat_dst   = ws + o; o += (size_t)N_NODES * HD;
    float* feat_src   = ws + o; o += (size_t)R_REL * N_NODES * HD;
    float* aggb       = ws + o; o += (size_t)R_REL * N_NODES * HD;
    float* ebuf       = ws + o; o += (size_t)R_REL * E_EDGES * H_HEADS;
    float* mbuf       = ws + o; o += (size_t)R_REL * N_NODES * H_HEADS;
    float* sbuf       = ws + o; o += (size_t)R_REL * N_NODES * H_HEADS;
    float* esb        = ws + o; o += (size_t)R_REL * N_NODES * H_HEADS;
    float* edb        = ws + o; o += (size_t)R_REL * N_NODES * H_HEADS;
    float* scoresb    = ws + o; o += (size_t)R_REL * N_NODES * H_HEADS;
    float* attnb      = ws + o; o += R_REL * H_HEADS * 2 * D_HID;   // 512
    float* relfeat    = ws + o; o += R_REL * H_HEADS * D_HID;       // 256

    const int B = 256;
    const int totalE = R_REL * E_EDGES;

    // ---- init accumulators (fresh every call: graph-replay safe) ----
    fill_f32<<<cdiv_ll(2LL * N_NODES, B), B, 0, stream>>>(deg_out, 0.0f, 2LL * N_NODES);
    fill_f32<<<cdiv_ll((long long)N_NODES * HD, B), B, 0, stream>>>(hg, 0.0f, (long long)N_NODES * HD);
    fill_f32<<<cdiv_ll((long long)R_REL * N_NODES * HD, B), B, 0, stream>>>(aggb, 0.0f, (long long)R_REL * N_NODES * HD);
    fill_f32<<<cdiv_ll((long long)R_REL * N_NODES * H_HEADS, B), B, 0, stream>>>(mbuf, -FLT_MAX, (long long)R_REL * N_NODES * H_HEADS);
    fill_f32<<<cdiv_ll((long long)R_REL * N_NODES * H_HEADS, B), B, 0, stream>>>(sbuf, 0.0f, (long long)R_REL * N_NODES * H_HEADS);

    // ---- degrees + GraphConv scatter ----
    degree_kernel<<<cdiv_ll(totalE, B), B, 0, stream>>>(src, dst, deg_out, deg_in, totalE);
    gcn_scatter<<<cdiv_ll((long long)totalE * HD, B), B, 0, stream>>>(x, src, dst, deg_out, hg, totalE);

    // ---- dense GEMMs via f32 WMMA ----
    const int gemm_grid = N_NODES / 16;   // 6250
    gemm128_wmma<<<gemm_grid, 256, 0, stream>>>(hg, W_g, b_g, deg_in, node_level, /*relu=*/1);
    gemm128_wmma<<<gemm_grid, 256, 0, stream>>>(x, W_node, b_node, nullptr, feat_dst, 0);
    gemm128_wmma<<<gemm_grid, 256, 0, stream>>>(x, W_src, b_src, nullptr, feat_src, 0);
    gemm128_wmma<<<gemm_grid, 256, 0, stream>>>(x, W_src + (size_t)DIN * HD, b_src + HD, nullptr,
                                                feat_src + (size_t)N_NODES * HD, 0);

    // ---- tiny relation tensors ----
    small_precompute<<<1, 256, 0, stream>>>(rel_emb, W_relT, W_relprop, b_relprop, W_fuse, attnb, relfeat);

    // ---- attention: per-node scores, then 3 edge passes ----
    node_scores<<<cdiv_ll((long long)R_REL * N_NODES * H_HEADS, B), B, 0, stream>>>(feat_src, feat_dst, attnb, esb, edb);
    edge_max<<<cdiv_ll(totalE, B), B, 0, stream>>>(src, dst, esb, edb, ebuf, mbuf);
    edge_expsum<<<cdiv_ll(totalE, B), B, 0, stream>>>(dst, mbuf, ebuf, sbuf);
    edge_agg<<<cdiv_ll((long long)totalE * HD, B), B, 0, stream>>>(src, dst, ebuf, sbuf, feat_src, aggb);

    // ---- fusion ----
    rel_scores<<<cdiv_ll((long long)R_REL * N_NODES * H_HEADS, B), B, 0, stream>>>(aggb, relfeat, scoresb);
    final_fuse<<<cdiv_ll((long long)N_NODES * HD, B), B, 0, stream>>>(aggb, scoresb, feat_dst, node_level,
                                                                      residual_w, scale_w, out);
    (void)in_sizes; (void)n_in; (void)out_size; (void)ws_size;
}